// Model_20942260535431
// MI455X (gfx1250) — compile-verified
//
#include <hip/hip_runtime.h>
#include <math.h>

#define Bn 16384
#define Tn 48
#define Pn 13
#define Hn 50
#define Cn 5

// ---- LDS layout (floats). One array => staging stores may-alias frag loads,
// keeping weight reads in-loop (ds_load) instead of hoisted+spilled.
#define FGH 0        // 16 frags  [nt*4+ks]           W_gh.T   (N=50pad64, K=13pad16)
#define FHIST 1024   // 13 frags  [ks]                W_hist.T (N=13pad16, K=50pad52)
#define FFEAT 1856   // 4 frags   [ks]                Wf.T     (zero diag)
#define FCOMB 2112   // 7 frags   [ks]                W_comb.T (K=26pad28)
#define FIH 2560     // 112 frags [(gi*4+nt)*7+ks]    W_ih.T gate blocks
#define FHH 9728     // 208 frags [(gi*4+nt)*13+ks]   W_hh.T gate blocks
#define FRAG_TOTAL 23040
#define STAGE_SZ 2240  // per wave
#define SB_D 0         // 16x16
#define SB_XC 256      // 16x16
#define SB_A 512       // 16x28
#define SB_I 960       // 16x28
#define SB_H 1408      // 16x52
#define SMEM_FLOATS (FRAG_TOTAL + 4 * STAGE_SZ)  // 32000 floats = 128000 B

typedef float v2f __attribute__((ext_vector_type(2)));
typedef float v8f __attribute__((ext_vector_type(8)));

// ---------------- WMMA helper: D = A(16x4,f32) * B(4x16,f32) + C ----------------
static __device__ __forceinline__ v8f wmma4(v2f a, v2f b, v8f c) {
    return __builtin_amdgcn_wmma_f32_16x16x4_f32(false, a, false, b, (short)0, c,
                                                 false, false);
}

static __device__ __forceinline__ v8f zero8() {
    v8f r;
#pragma unroll
    for (int i = 0; i < 8; ++i) r[i] = 0.f;
    return r;
}

static __device__ __forceinline__ float sigf(float x) {
    return 1.f / (1.f + __expf(-x));
}

// Intra-wave staging fence (per-wave buffers; DS is in-order per wave in HW)
static __device__ __forceinline__ void wavefence() {
#if __has_builtin(__builtin_amdgcn_wave_barrier) && __has_builtin(__builtin_amdgcn_fence)
    __builtin_amdgcn_fence(__ATOMIC_RELEASE, "workgroup");
    __builtin_amdgcn_wave_barrier();
    __builtin_amdgcn_fence(__ATOMIC_ACQUIRE, "workgroup");
#else
    __syncthreads();
#endif
}

// A-fragment (16x4 f32) from LDS tile stored row-major [16][LD].
static __device__ __forceinline__ v2f ldsA(const float* buf, int LD, int k,
                                           int lcol, int lhalf) {
    int idx = lcol * LD + k + (lhalf << 1);
    v2f r;
    r.x = buf[idx];
    r.y = buf[idx + 1];
    return r;
}

// B-fragment from fragment-linear LDS: frag = 64 consecutive floats,
// lane l reads dwords 2l,2l+1 -> single ds_load_b64, conflict-free.
static __device__ __forceinline__ v2f ldsB(const float* fr, int fragIdx, int lane) {
    const float* p = fr + (fragIdx << 6) + (lane << 1);
    v2f r;
    r.x = p[0];
    r.y = p[1];
    return r;
}

// Bias broadcast tile in C/D layout (branchless clamp+mask)
static __device__ __forceinline__ v8f bTile(const float* b, int n0, int nmax,
                                            int lcol) {
    int cidx = n0 + lcol;
    int cc = (cidx < nmax) ? cidx : (nmax - 1);
    float val = b[cc] * ((cidx < nmax) ? 1.f : 0.f);
    v8f r;
#pragma unroll
    for (int v = 0; v < 8; ++v) r[v] = val;
    return r;
}

static __device__ __forceinline__ v8f gateBias(const float* bi, const float* bh,
                                               int gi, int nl) {
    int nlc = (nl < Hn) ? nl : (Hn - 1);
    int n = gi * Hn + nlc;
    float val = (bi[n] + bh[n]) * ((nl < Hn) ? 1.f : 0.f);
    v8f r;
#pragma unroll
    for (int v = 0; v < 8; ++v) r[v] = val;
    return r;
}

// Load a [16 rows, 16 cols] C/D-layout tile of src[B,T,P] at time t (cols>=P -> 0)
static __device__ __forceinline__ v8f loadCD(const float* src, int b0, int t,
                                             int lcol, int lhalf) {
    float valid = (lcol < Pn) ? 1.f : 0.f;
    int p = (lcol < Pn) ? lcol : 0;
    v8f r;
#pragma unroll
    for (int v = 0; v < 8; ++v) {
        int row = b0 + lhalf * 8 + v;
        r[v] = src[(row * Tn + t) * Pn + p] * valid;
    }
    return r;
}

// Store C/D-layout tile into row-major LDS [16][LD] at column offset col0
static __device__ __forceinline__ void stageTile(float* buf, int LD, v8f tv,
                                                 int col0, int colLimit,
                                                 int lcol, int lhalf) {
    int col = col0 + lcol;
    if (col < colLimit) {
        int rb = lhalf << 3;
#pragma unroll
        for (int v = 0; v < 8; ++v) buf[(rb + v) * LD + col] = tv[v];
    }
}

// ---------------- prologue fills: global weights -> fragment-linear LDS --------
// element order inside a frag: r = lane*2 + comp
static __device__ void fillB(float* dst, const float* W, int nTiles, int kSteps,
                             int nmax, int K, int tid) {
    int total = nTiles * kSteps * 64;
    for (int i = tid; i < total; i += 128) {
        int f = i >> 6, r = i & 63;
        int ks = f % kSteps, nt = f / kSteps;
        int lane = r >> 1, comp = r & 1;
        int n = nt * 16 + (lane & 15);
        int k = ks * 4 + ((lane >> 4) << 1) + comp;
        dst[i] = (n < nmax && k < K) ? W[n * K + k] : 0.f;
    }
}

static __device__ void fillBfeat(float* dst, const float* Wf, int tid) {
    for (int i = tid; i < 4 * 64; i += 128) {
        int f = i >> 6, r = i & 63;
        int lane = r >> 1, comp = r & 1;
        int n = lane & 15;
        int k = f * 4 + ((lane >> 4) << 1) + comp;
        dst[i] = (n < Pn && k < Pn && n != k) ? Wf[n * Pn + k] : 0.f;
    }
}

static __device__ void fillBgate(float* dst, const float* W, int kSteps, int K,
                                 int tid) {
    int total = 16 * kSteps * 64;  // (gi*4+nt) x kSteps frags
    for (int i = tid; i < total; i += 128) {
        int f = i >> 6, r = i & 63;
        int ks = f % kSteps;
        int gnt = f / kSteps;
        int gi = gnt >> 2, nt = gnt & 3;
        int lane = r >> 1, comp = r & 1;
        int nl = nt * 16 + (lane & 15);
        int k = ks * 4 + ((lane >> 4) << 1) + comp;
        dst[i] = (nl < Hn && k < K) ? W[(gi * Hn + nl) * K + k] : 0.f;
    }
}

// ---------------- kernels ----------------
__global__ void init_kernel(float* ws) {
    if (threadIdx.x < 2) ws[threadIdx.x] = 0.f;
}

__global__ void msum_kernel(const float* __restrict__ masks, float* ws) {
    int t = blockIdx.x;
    float s = 0.f;
    for (int i = threadIdx.x; i < Bn * Pn; i += 256) {
        int b = i / Pn;
        int p = i - b * Pn;
        s += masks[(b * Tn + t) * Pn + p];
    }
    __shared__ float red[256];
    red[threadIdx.x] = s;
    __syncthreads();
    for (int off = 128; off > 0; off >>= 1) {
        if (threadIdx.x < (unsigned)off) red[threadIdx.x] += red[threadIdx.x + off];
        __syncthreads();
    }
    if (threadIdx.x == 0) ws[2 + t] = red[0] + 1e-5f;
}

__global__ __launch_bounds__(128) void scan_kernel(
    const float* __restrict__ values, const float* __restrict__ masks,
    const float* __restrict__ deltas, const float* __restrict__ labels,
    const float* __restrict__ W_gh, const float* __restrict__ b_gh,
    const float* __restrict__ w_gx, const float* __restrict__ b_gx,
    const float* __restrict__ W_hist, const float* __restrict__ b_hist,
    const float* __restrict__ W_feat, const float* __restrict__ b_feat,
    const float* __restrict__ W_comb, const float* __restrict__ b_comb,
    const float* __restrict__ W_ih, const float* __restrict__ W_hh,
    const float* __restrict__ b_ih, const float* __restrict__ b_hh,
    const float* __restrict__ W_out, const float* __restrict__ b_out,
    const float* __restrict__ msum, float* acc) {
    __shared__ float smem[SMEM_FLOATS];  // 92KB frags + 4x staging (320KB WGP LDS)

    const int tid = threadIdx.x;
    const int wv = tid >> 5;
    const int lane = tid & 31;
    const int lcol = lane & 15;
    const int lhalf = lane >> 4;
    const int b0 = (blockIdx.x * 4 + wv) * 16;

    // ---- one-time: weights -> fragment-linear LDS ----
    fillB(smem + FGH, W_gh, 4, 4, Hn, Pn, tid);
    fillB(smem + FHIST, W_hist, 1, 13, Pn, Hn, tid);
    fillBfeat(smem + FFEAT, W_feat, tid);
    fillB(smem + FCOMB, W_comb, 1, 7, Pn, 2 * Pn, tid);
    fillBgate(smem + FIH, W_ih, 7, 2 * Pn, tid);
    fillBgate(smem + FHH, W_hh, 13, Hn, tid);
    __syncthreads();

    float* stg = smem + FRAG_TOTAL + wv * STAGE_SZ;
    float* dbuf = stg + SB_D;
    float* xcbuf = stg + SB_XC;
    float* abuf = stg + SB_A;
    float* ibuf = stg + SB_I;
    float* hbuf = stg + SB_H;

    v8f h[4], c[4];
#pragma unroll
    for (int nt = 0; nt < 4; ++nt) {
        h[nt] = zero8();
        c[nt] = zero8();
    }
    float xl = 0.f;

    for (int t = 0; t < Tn; ++t) {
        // keep weight/bias reads in-loop (LDS/L0-resident) instead of hoisted
        asm volatile("" ::: "memory");

        v8f X = loadCD(values, b0, t, lcol, lhalf);
        v8f Mk = loadCD(masks, b0, t, lcol, lhalf);
        v8f Dl = loadCD(deltas, b0, t, lcol, lhalf);
        float inv_ms = 1.f / msum[t];

        if (t + 1 < Tn) {  // gfx1250 global_prefetch_b8 for next timestep
            int row = b0 + lhalf * 8;
            __builtin_prefetch(&values[(row * Tn + t + 1) * Pn], 0, 0);
            __builtin_prefetch(&masks[(row * Tn + t + 1) * Pn], 0, 0);
            __builtin_prefetch(&deltas[(row * Tn + t + 1) * Pn], 0, 0);
        }

        // ---- gamma_h = exp(-relu(d @ W_gh.T + b_gh)); h *= gamma_h ----
        stageTile(dbuf, 16, Dl, 0, 16, lcol, lhalf);
        wavefence();
        v2f dA[4];
#pragma unroll
        for (int ks = 0; ks < 4; ++ks) dA[ks] = ldsA(dbuf, 16, ks * 4, lcol, lhalf);
#pragma unroll
        for (int nt = 0; nt < 4; ++nt) {
            v8f a = bTile(b_gh, nt * 16, Hn, lcol);
#pragma unroll
            for (int ks = 0; ks < 4; ++ks)
                a = wmma4(dA[ks], ldsB(smem + FGH, nt * 4 + ks, lane), a);
#pragma unroll
            for (int v = 0; v < 8; ++v) h[nt][v] *= __expf(-fmaxf(a[v], 0.f));
        }

        // ---- gamma_x = exp(-relu(d * w_gx + b_gx)) (diag decay) ----
        v8f GX;
        {
            int p = (lcol < Pn) ? lcol : 0;
            float wg = w_gx[p], bg = b_gx[p];
#pragma unroll
            for (int v = 0; v < 8; ++v)
                GX[v] = __expf(-fmaxf(Dl[v] * wg + bg, 0.f));
        }

        // ---- stage decayed h; build 13 shared A-fragments (K=50->52) ----
#pragma unroll
        for (int nt = 0; nt < 4; ++nt)
            stageTile(hbuf, 52, h[nt], nt * 16, 52, lcol, lhalf);
        wavefence();
        v2f hA[13];
#pragma unroll
        for (int ks = 0; ks < 13; ++ks)
            hA[ks] = ldsA(hbuf, 52, ks * 4, lcol, lhalf);

        // ---- x_h = h @ W_hist.T + b_hist ----
        v8f XH = bTile(b_hist, 0, Pn, lcol);
#pragma unroll
        for (int ks = 0; ks < 13; ++ks)
            XH = wmma4(hA[ks], ldsB(smem + FHIST, ks, lane), XH);

        float s = 0.f;
        v8f XC;
#pragma unroll
        for (int v = 0; v < 8; ++v) {
            s += fabsf(X[v] - XH[v]) * Mk[v];
            XC[v] = Mk[v] * X[v] + (1.f - Mk[v]) * XH[v];
        }

        // ---- z_h = x_c @ Wf.T + b_feat (zero-diag Wf) ----
        stageTile(xcbuf, 16, XC, 0, 16, lcol, lhalf);
        wavefence();
        v2f xcA[4];
#pragma unroll
        for (int ks = 0; ks < 4; ++ks)
            xcA[ks] = ldsA(xcbuf, 16, ks * 4, lcol, lhalf);
        v8f ZH = bTile(b_feat, 0, Pn, lcol);
#pragma unroll
        for (int ks = 0; ks < 4; ++ks)
            ZH = wmma4(xcA[ks], ldsB(smem + FFEAT, ks, lane), ZH);
#pragma unroll
        for (int v = 0; v < 8; ++v) s += fabsf(X[v] - ZH[v]) * Mk[v];

        // ---- alpha = [gamma_x | m] @ W_comb.T + b_comb ----
        stageTile(abuf, 28, GX, 0, Pn, lcol, lhalf);
        stageTile(abuf, 28, Mk, Pn, 2 * Pn, lcol, lhalf);
        if (lcol < 2) {
            int rb = lhalf << 3;
#pragma unroll
            for (int v = 0; v < 8; ++v) abuf[(rb + v) * 28 + 26 + lcol] = 0.f;
        }
        wavefence();
        v2f aA[7];
#pragma unroll
        for (int ks = 0; ks < 7; ++ks)
            aA[ks] = ldsA(abuf, 28, ks * 4, lcol, lhalf);
        v8f AL = bTile(b_comb, 0, Pn, lcol);
#pragma unroll
        for (int ks = 0; ks < 7; ++ks)
            AL = wmma4(aA[ks], ldsB(smem + FCOMB, ks, lane), AL);

        // ---- combine, loss, c_c ----
        v8f CC;
#pragma unroll
        for (int v = 0; v < 8; ++v) {
            float ch = AL[v] * ZH[v] + (1.f - AL[v]) * XH[v];
            s += fabsf(X[v] - ch) * Mk[v];
            CC[v] = Mk[v] * X[v] + (1.f - Mk[v]) * ch;
        }
        xl += s * inv_ms;

        // ---- inp = [c_c | m]; 7 A-fragments ----
        stageTile(ibuf, 28, CC, 0, Pn, lcol, lhalf);
        stageTile(ibuf, 28, Mk, Pn, 2 * Pn, lcol, lhalf);
        if (lcol < 2) {
            int rb = lhalf << 3;
#pragma unroll
            for (int v = 0; v < 8; ++v) ibuf[(rb + v) * 28 + 26 + lcol] = 0.f;
        }
        wavefence();
        v2f iA[7];
#pragma unroll
        for (int ks = 0; ks < 7; ++ks)
            iA[ks] = ldsA(ibuf, 28, ks * 4, lcol, lhalf);

        // ---- gates = inp @ W_ih.T + h @ W_hh.T + b_ih + b_hh; LSTM update ----
#pragma unroll
        for (int nt = 0; nt < 4; ++nt) {
            int nl = nt * 16 + lcol;
            v8f gt[4];
#pragma unroll
            for (int gi = 0; gi < 4; ++gi) {
                int gnt = gi * 4 + nt;
                v8f a = gateBias(b_ih, b_hh, gi, nl);
#pragma unroll
                for (int ks = 0; ks < 7; ++ks)
                    a = wmma4(iA[ks], ldsB(smem + FIH, gnt * 7 + ks, lane), a);
#pragma unroll
                for (int ks = 0; ks < 13; ++ks)
                    a = wmma4(hA[ks], ldsB(smem + FHH, gnt * 13 + ks, lane), a);
                gt[gi] = a;
            }
#pragma unroll
            for (int v = 0; v < 8; ++v) {
                float ig = sigf(gt[0][v]);
                float fg = sigf(gt[1][v]);
                float gg = tanhf(gt[2][v]);
                float og = sigf(gt[3][v]);
                float cn = fg * c[nt][v] + ig * gg;
                c[nt][v] = cn;
                h[nt][v] = og * tanhf(cn);
            }
        }
        wavefence();
    }

    // ---- classification head: y_h = sigmoid(h @ W_out.T + b_out); BCE(sig(y_h)) ----
#pragma unroll
    for (int nt = 0; nt < 4; ++nt)
        stageTile(hbuf, 52, h[nt], nt * 16, 52, lcol, lhalf);
    wavefence();
    float yl = 0.f;
    if (lhalf == 0) {
        int r = lcol;
        for (int cls = 0; cls < Cn; ++cls) {
            float pre = b_out[cls];
            for (int j = 0; j < Hn; ++j)
                pre += hbuf[r * 52 + j] * W_out[cls * Hn + j];
            float yh = sigf(pre);
            float lab = labels[(b0 + r) * Cn + cls];
            float lsp = -log1pf(__expf(-yh));  // log_sigmoid(yh)
            float lsn = -log1pf(__expf(yh));   // log_sigmoid(-yh)
            yl += -(lab * lsp + (1.f - lab) * lsn);
        }
    }
    for (int off = 16; off > 0; off >>= 1) {
        xl += __shfl_xor(xl, off, 32);
        yl += __shfl_xor(yl, off, 32);
    }
    if (lane == 0) {
        atomicAdd(&acc[0], xl);
        atomicAdd(&acc[1], yl);
    }
}

__global__ void finalize_kernel(const float* acc, const float* __restrict__ W_out,
                                float* out) {
    float l1 = 0.f, l2 = 0.f;
    for (int i = 0; i < Cn * Hn; ++i) {
        float w = W_out[i];
        l1 += fabsf(w);
        l2 += w * w;
    }
    l2 = sqrtf(l2);
    float xl = acc[0] / (float)Tn;
    float yl = acc[1] / (float)(Bn * Cn);
    yl += 1e-4f * (0.5f * (1.f - 0.9f) * l2 + 0.9f * l1);
    out[0] = xl * 1.0f + yl * 1.0f;
}

extern "C" void kernel_launch(void* const* d_in, const int* in_sizes, int n_in,
                              void* d_out, int out_size, void* d_ws, size_t ws_size,
                              hipStream_t stream) {
    const float* values = (const float*)d_in[0];
    const float* masks  = (const float*)d_in[1];
    const float* deltas = (const float*)d_in[2];
    const float* labels = (const float*)d_in[3];
    const float* W_gh   = (const float*)d_in[4];
    const float* b_gh   = (const float*)d_in[5];
    const float* w_gx   = (const float*)d_in[6];
    const float* b_gx   = (const float*)d_in[7];
    const float* W_hist = (const float*)d_in[8];
    const float* b_hist = (const float*)d_in[9];
    const float* W_feat = (const float*)d_in[10];
    const float* b_feat = (const float*)d_in[11];
    const float* W_comb = (const float*)d_in[12];
    const float* b_comb = (const float*)d_in[13];
    const float* W_ih   = (const float*)d_in[14];
    const float* W_hh   = (const float*)d_in[15];
    const float* b_ih   = (const float*)d_in[16];
    const float* b_hh   = (const float*)d_in[17];
    const float* W_out  = (const float*)d_in[18];
    const float* b_out  = (const float*)d_in[19];
    float* ws  = (float*)d_ws;   // ws[0]=x_loss acc, ws[1]=y_loss acc, ws[2..49]=msum[t]
    float* out = (float*)d_out;

    hipLaunchKernelGGL(init_kernel, dim3(1), dim3(32), 0, stream, ws);
    hipLaunchKernelGGL(msum_kernel, dim3(Tn), dim3(256), 0, stream, masks, ws);
    hipLaunchKernelGGL(scan_kernel, dim3(Bn / 64), dim3(128), 0, stream,
                       values, masks, deltas, labels, W_gh, b_gh, w_gx, b_gx,
                       W_hist, b_hist, W_feat, b_feat, W_comb, b_comb,
                       W_ih, W_hh, b_ih, b_hh, W_out, b_out,
                       ws + 2, ws);
    hipLaunchKernelGGL(finalize_kernel, dim3(1), dim3(1), 0, stream, ws, W_out, out);
}